// net_gcn_20547123544188
// MI455X (gfx1250) — compile-verified
//
#include <hip/hip_runtime.h>
#include <hip/hip_bf16.h>

#define N_NODES 50000
#define N_EDGES 1600000
#define EPS 1e-5f

typedef __attribute__((ext_vector_type(2))) float v2f;
typedef __attribute__((ext_vector_type(8))) float v8f;

// ---------------------------------------------------------------------------
// Degree / norm helpers
// ---------------------------------------------------------------------------
__global__ void k_init_deg(float* __restrict__ deg) {
    int i = blockIdx.x * 256 + threadIdx.x;
    if (i < N_NODES) deg[i] = 1.0f;  // self-loop contributes 1
}

__global__ void k_zero(float* __restrict__ p, int n) {
    int i = blockIdx.x * 256 + threadIdx.x;
    if (i < n) p[i] = 0.0f;
}

__global__ void k_deg_edges(const int* __restrict__ dst, float* __restrict__ deg) {
    int e = blockIdx.x * 256 + threadIdx.x;
    if (e < N_EDGES) unsafeAtomicAdd(&deg[dst[e]], 1.0f);
}

__global__ void k_rsqrt_inplace(float* __restrict__ deg) {
    int i = blockIdx.x * 256 + threadIdx.x;
    if (i < N_NODES) deg[i] = rsqrtf(deg[i]);   // deg -> dinv
}

// ---------------------------------------------------------------------------
// BatchNorm statistics (column sums / sums of squares), fully coalesced
// ---------------------------------------------------------------------------
template <int C>
__global__ void k_bn_stats(const float* __restrict__ x,
                           float* __restrict__ sums, float* __restrict__ sumsq) {
    constexpr int PER = C / 256;
    float s[PER], q[PER];
#pragma unroll
    for (int j = 0; j < PER; ++j) { s[j] = 0.f; q[j] = 0.f; }
    for (int r = blockIdx.x; r < N_NODES; r += gridDim.x) {
        const float* row = x + (size_t)r * C;
#pragma unroll
        for (int j = 0; j < PER; ++j) {
            float v = row[threadIdx.x + 256 * j];
            s[j] += v; q[j] += v * v;
        }
    }
#pragma unroll
    for (int j = 0; j < PER; ++j) {
        unsafeAtomicAdd(&sums[threadIdx.x + 256 * j], s[j]);
        unsafeAtomicAdd(&sumsq[threadIdx.x + 256 * j], q[j]);
    }
}

// avec = gamma * rsqrt(var+eps), cvec = beta - mean*avec
template <int C>
__global__ void k_bn_finalize(const float* __restrict__ sums, const float* __restrict__ sumsq,
                              const float* __restrict__ gamma, const float* __restrict__ beta,
                              float* __restrict__ avec, float* __restrict__ cvec) {
    int c = blockIdx.x * 256 + threadIdx.x;
    if (c < C) {
        const float inv_n = 1.0f / (float)N_NODES;
        float mu  = sums[c] * inv_n;
        float var = sumsq[c] * inv_n - mu * mu;
        float a   = gamma[c] * rsqrtf(var + EPS);
        avec[c] = a;
        cvec[c] = beta[c] - mu * a;
    }
}

// Wp[k][j] = avec[k] * W[k][j]   (N fixed at 256)
__global__ void k_fuse_w(const float* __restrict__ W, const float* __restrict__ avec,
                         float* __restrict__ Wp) {
    int k = blockIdx.x, j = threadIdx.x;
    Wp[k * 256 + j] = avec[k] * W[k * 256 + j];
}

// biasadd[j] = sum_k cvec[k] * W[k][j]
__global__ void k_bias_fold(const float* __restrict__ W, const float* __restrict__ cvec,
                            float* __restrict__ biasadd, int K) {
    int j = threadIdx.x;
    float acc = 0.f;
    for (int k = 0; k < K; ++k) acc += cvec[k] * W[k * 256 + j];
    biasadd[j] = acc;
}

// ---------------------------------------------------------------------------
// GEMM:  C[M x 256] = A[M x K] @ B[K x 256] + bias   (fp32 WMMA 16x16x4)
// Block = 256 threads = 8 waves; block computes a 32-row x 256-col stripe.
// Wave w owns columns [32w, 32w+32) and both 16-row M subtiles -> 2x2 grid of
// 16x16 WMMA accumulators; each B fragment is reused for 2 WMMAs.
// A tile (64K x 32M) staged in LDS, k-major, pad-33 stride (conflict free).
// ---------------------------------------------------------------------------
template <int K>
__launch_bounds__(256)
__global__ void k_gemm_wmma(const float* __restrict__ A, const float* __restrict__ B,
                            const float* __restrict__ bias, float* __restrict__ Cout) {
    __shared__ float lds[64 * 33];
    const int tid  = threadIdx.x;
    const int row0 = blockIdx.x * 32;
    const int lane = tid & 31;
    const int w    = tid >> 5;
    const int lg   = lane >> 4;   // lane group: 0 -> K pair (k,k+1), 1 -> (k+2,k+3)
    const int l    = lane & 15;
    const int c0   = w * 32;

    v8f acc00 = {0.f,0.f,0.f,0.f,0.f,0.f,0.f,0.f};
    v8f acc01 = acc00, acc10 = acc00, acc11 = acc00;

    for (int kk = 0; kk < K; kk += 64) {
        __syncthreads();
#pragma unroll
        for (int it = 0; it < 8; ++it) {      // 32 rows x 64 cols, coalesced
            int idx = it * 256 + tid;
            int r = idx >> 6, c = idx & 63;
            int row = row0 + r;
            lds[c * 33 + r] = (row < N_NODES) ? A[(size_t)row * K + kk + c] : 0.f;
        }
        // prefetch next K chunk of A while we compute on this one
        if (kk + 64 < K && tid < 32) {
            int row = row0 + tid;
            if (row < N_NODES) {
                const float* p = &A[(size_t)row * K + kk + 64];
                __builtin_prefetch(p, 0, 3);
                __builtin_prefetch(p + 32, 0, 3);
            }
        }
        __syncthreads();
#pragma unroll
        for (int k = 0; k < 64; k += 4) {
            const int kb = k + 2 * lg;
            v2f a0, a1;
            a0.x = lds[kb * 33 + l];            a0.y = lds[(kb + 1) * 33 + l];
            a1.x = lds[kb * 33 + 16 + l];       a1.y = lds[(kb + 1) * 33 + 16 + l];
            const float* Bp = B + (size_t)(kk + kb) * 256;
            v2f b0; b0.x = Bp[c0 + l];      b0.y = Bp[256 + c0 + l];
            v2f b1; b1.x = Bp[c0 + 16 + l]; b1.y = Bp[256 + c0 + 16 + l];
            acc00 = __builtin_amdgcn_wmma_f32_16x16x4_f32(false, a0, false, b0,
                                                          (short)0, acc00, false, false);
            acc01 = __builtin_amdgcn_wmma_f32_16x16x4_f32(false, a0, false, b1,
                                                          (short)0, acc01, false, false);
            acc10 = __builtin_amdgcn_wmma_f32_16x16x4_f32(false, a1, false, b0,
                                                          (short)0, acc10, false, false);
            acc11 = __builtin_amdgcn_wmma_f32_16x16x4_f32(false, a1, false, b1,
                                                          (short)0, acc11, false, false);
        }
    }
    const float bia0 = bias[c0 + l];
    const float bia1 = bias[c0 + 16 + l];
#pragma unroll
    for (int r = 0; r < 8; ++r) {             // C layout: VGPR r -> M = r + 8*lg, N = l
        int row = row0 + r + 8 * lg;
        if (row < N_NODES) {
            Cout[(size_t)row * 256 + c0 + l]      = acc00[r] + bia0;
            Cout[(size_t)row * 256 + c0 + 16 + l] = acc01[r] + bia1;
        }
        int row2 = row + 16;
        if (row2 < N_NODES) {
            Cout[(size_t)row2 * 256 + c0 + l]      = acc10[r] + bia0;
            Cout[(size_t)row2 * 256 + c0 + 16 + l] = acc11[r] + bia1;
        }
    }
}

// ---------------------------------------------------------------------------
// Graph aggregation
// ---------------------------------------------------------------------------
// agg[i][:] = h[i][:] * dinv[i]^2    (self loop term, also initializes agg)
__global__ void k_self_init(const float* __restrict__ h, const float* __restrict__ dinv,
                            float* __restrict__ agg) {
    int i = blockIdx.x * 256 + threadIdx.x;   // grid covers N_NODES*256 exactly
    int node = i >> 8;
    float d = dinv[node];
    agg[i] = h[i] * d * d;
}

// agg[dst][:] += h[src][:] * dinv[src]*dinv[dst]   (64 threads / edge, float4)
__global__ void k_edge_scatter(const int* __restrict__ src, const int* __restrict__ dst,
                               const float* __restrict__ dinv, const float* __restrict__ h,
                               float* __restrict__ agg) {
    int e = blockIdx.x * 4 + (threadIdx.x >> 6);
    if (e >= N_EDGES) return;
    int s = src[e], d = dst[e];
    float wgt = dinv[s] * dinv[d];
    int f = (threadIdx.x & 63) * 4;
    const float4 hv = *reinterpret_cast<const float4*>(h + (size_t)s * 256 + f);
    float* out = agg + (size_t)d * 256 + f;
    unsafeAtomicAdd(out + 0, hv.x * wgt);
    unsafeAtomicAdd(out + 1, hv.y * wgt);
    unsafeAtomicAdd(out + 2, hv.z * wgt);
    unsafeAtomicAdd(out + 3, hv.w * wgt);
}

// in-place: v = relu(v + b[j])
__global__ void k_bias_relu(float* __restrict__ agg, const float* __restrict__ b) {
    int i = blockIdx.x * 256 + threadIdx.x;
    float v = agg[i] + b[i & 255];
    agg[i] = v > 0.f ? v : 0.f;
}

// ---------------------------------------------------------------------------
extern "C" void kernel_launch(void* const* d_in, const int* in_sizes, int n_in,
                              void* d_out, int out_size, void* d_ws, size_t ws_size,
                              hipStream_t stream) {
    (void)in_sizes; (void)n_in; (void)out_size; (void)ws_size;

    const float* x      = (const float*)d_in[0];   // [50000,512]
    const int*   src    = (const int*)d_in[1];     // edge_index[0]
    const int*   dst    = src + N_EDGES;           // edge_index[1]
    const float* gamma0 = (const float*)d_in[2];
    const float* beta0  = (const float*)d_in[3];
    const float* W0     = (const float*)d_in[4];   // [512,256]
    const float* b0     = (const float*)d_in[5];
    const float* gamma1 = (const float*)d_in[6];
    const float* beta1  = (const float*)d_in[7];
    const float* W1     = (const float*)d_in[8];   // [256,256]
    const float* b1     = (const float*)d_in[9];
    float* out = (float*)d_out;                    // [50000,256]

    // workspace layout (floats)
    float* ws      = (float*)d_ws;
    float* deg     = ws;                       // 50048 (deg -> dinv in place)
    float* hbuf    = deg    + 50048;           // 50000*256
    float* aggbuf  = hbuf   + 12800000;        // 50000*256
    float* W0p     = aggbuf + 12800000;        // 512*256
    float* W1p     = W0p    + 131072;          // 256*256
    float* sums    = W1p    + 65536;           // 512
    float* sumsq   = sums   + 512;             // 512
    float* avec    = sumsq  + 512;             // 512
    float* cvec    = avec   + 512;             // 512
    float* biasadd = cvec   + 512;             // 256

    const int FEAT_GRID = (N_NODES * 256) / 256;        // 50000
    const int EDGE_GRID = (N_EDGES + 3) / 4;            // 400000
    const int GEMM_GRID = (N_NODES + 31) / 32;          // 1563

    // ---- degrees & symmetric norm ----
    k_init_deg<<<(N_NODES + 255) / 256, 256, 0, stream>>>(deg);
    k_zero<<<4, 256, 0, stream>>>(sums, 1024);          // sums + sumsq
    k_deg_edges<<<(N_EDGES + 255) / 256, 256, 0, stream>>>(dst, deg);
    k_rsqrt_inplace<<<(N_NODES + 255) / 256, 256, 0, stream>>>(deg);  // deg -> dinv

    // ---- layer 0: BN stats -> fold into W0 ----
    k_bn_stats<512><<<512, 256, 0, stream>>>(x, sums, sumsq);
    k_bn_finalize<512><<<2, 256, 0, stream>>>(sums, sumsq, gamma0, beta0, avec, cvec);
    k_fuse_w<<<512, 256, 0, stream>>>(W0, avec, W0p);
    k_bias_fold<<<1, 256, 0, stream>>>(W0, cvec, biasadd, 512);

    // ---- GEMM1: h = BN(x) @ W0  (WMMA fp32) ----
    k_gemm_wmma<512><<<GEMM_GRID, 256, 0, stream>>>(x, W0p, biasadd, hbuf);

    // ---- aggregate 1 + bias + relu ----
    k_self_init<<<FEAT_GRID, 256, 0, stream>>>(hbuf, deg, aggbuf);
    k_edge_scatter<<<EDGE_GRID, 256, 0, stream>>>(src, dst, deg, hbuf, aggbuf);
    k_bias_relu<<<FEAT_GRID, 256, 0, stream>>>(aggbuf, b0);

    // ---- layer 1: BN stats -> fold into W1 ----
    k_zero<<<4, 256, 0, stream>>>(sums, 1024);
    k_bn_stats<256><<<512, 256, 0, stream>>>(aggbuf, sums, sumsq);
    k_bn_finalize<256><<<1, 256, 0, stream>>>(sums, sumsq, gamma1, beta1, avec, cvec);
    k_fuse_w<<<256, 256, 0, stream>>>(W1, avec, W1p);
    k_bias_fold<<<1, 256, 0, stream>>>(W1, cvec, biasadd, 256);

    // ---- GEMM2: h2 = BN(x1) @ W1 (reuse hbuf) ----
    k_gemm_wmma<256><<<GEMM_GRID, 256, 0, stream>>>(aggbuf, W1p, biasadd, hbuf);

    // ---- aggregate 2 directly into d_out + bias + relu ----
    k_self_init<<<FEAT_GRID, 256, 0, stream>>>(hbuf, deg, out);
    k_edge_scatter<<<EDGE_GRID, 256, 0, stream>>>(src, dst, deg, hbuf, out);
    k_bias_relu<<<FEAT_GRID, 256, 0, stream>>>(out, b1);
}